// GaussianPosterior_42150809043689
// MI455X (gfx1250) — compile-verified
//
#include <hip/hip_runtime.h>
#include <math.h>

typedef float v2f __attribute__((ext_vector_type(2)));
typedef float v8f __attribute__((ext_vector_type(8)));

#define DIM     512
#define KPROTO  128
#define NTILE   8          // KPROTO / 16 column tiles
#define TEMP_INV 10.0f     // 1 / TEMP
#define PULL     0.1f
#define EPSN     1e-12f

__device__ __forceinline__ float rcp_nr(float v) {
    // hardware rcp + one Newton-Raphson step (~1 ulp)
    float r = __builtin_amdgcn_rcpf(v);
    return r * (2.0f - v * r);
}

// ---------------------------------------------------------------------------
// Kernel 1: per-prototype constants
//   c[k]     = sum_d ( log(2*pi*v) + m^2 / v )
//   mnorm[k] = || m_k ||_2
// ---------------------------------------------------------------------------
__global__ __launch_bounds__(32)
void proto_stats_kernel(const float* __restrict__ means,
                        const float* __restrict__ vars,
                        float* __restrict__ c_out,
                        float* __restrict__ mnorm_out) {
    const int k    = blockIdx.x;
    const int lane = threadIdx.x;
    const float* m = means + (size_t)k * DIM;
    const float* v = vars  + (size_t)k * DIM;
    float csum = 0.0f, msq = 0.0f;
    for (int d = lane; d < DIM; d += 32) {
        float mm = m[d], vv = v[d];
        csum += logf(6.283185307179586f * vv) + mm * mm / vv;
        msq  += mm * mm;
    }
#pragma unroll
    for (int off = 16; off > 0; off >>= 1) {
        csum += __shfl_xor(csum, off, 32);
        msq  += __shfl_xor(msq,  off, 32);
    }
    if (lane == 0) {
        c_out[k]     = csum;
        mnorm_out[k] = sqrtf(msq);
    }
}

// ---------------------------------------------------------------------------
// Kernel 2: main fused kernel. One wave32 per 16-row tile of A.
//   S[n,k]  = sum_d a^2 * (-0.5/v)  +  a * (m/v)      (v_wmma_f32_16x16x4_f32)
//   G3[n,k] = sum_d a * m
//   logp    = S - 0.5*c_k ; softmax(logp/TEMP) max + argmax ; cosine ; partial
// Inner loop is software-pipelined: all 17 b64 loads of chunk d+4 are issued
// before the 24 WMMAs of chunk d execute, keeping L2 latency hidden.
// ---------------------------------------------------------------------------
__global__ __launch_bounds__(32)
void gauss_posterior_main(const float* __restrict__ a,
                          const float* __restrict__ means,
                          const float* __restrict__ vars,
                          const float* __restrict__ c_in,
                          const float* __restrict__ mnorm_in,
                          float* __restrict__ partial,
                          int N) {
    const int lane = threadIdx.x;
    const int m0   = blockIdx.x * 16;
    const int col  = lane & 15;   // sample row for A-frag; proto row for B-frags
    const int hi   = lane >> 4;   // half-wave select -> inner-K offset {0,2}

    const float* aP = a     + (size_t)(m0 + col) * DIM + hi * 2;
    const float* vP = vars  + (size_t)col * DIM + hi * 2;
    const float* mP = means + (size_t)col * DIM + hi * 2;

    v8f accLP[NTILE];   // S accumulators (log-prob GEMM)
    v8f accG3[NTILE];   // a . m accumulators
    const v8f vz = {0.f,0.f,0.f,0.f,0.f,0.f,0.f,0.f};
#pragma unroll
    for (int t = 0; t < NTILE; ++t) { accLP[t] = vz; accG3[t] = vz; }

    float nsq = 0.0f;   // per-lane partial of ||a_row||^2

    // ---- software pipeline: prologue loads for chunk 0 ----
    v2f avB, vvB[NTILE], mmB[NTILE];
    avB = *(const v2f*)(aP);
#pragma unroll
    for (int t = 0; t < NTILE; ++t) {
        vvB[t] = *(const v2f*)(vP + t * (16 * DIM));
        mmB[t] = *(const v2f*)(mP + t * (16 * DIM));
    }

#pragma unroll 1
    for (int d = 0; d < DIM - 4; d += 4) {
        // grab current chunk from buffers
        v2f av = avB;
        v2f vv[NTILE], mm[NTILE];
#pragma unroll
        for (int t = 0; t < NTILE; ++t) { vv[t] = vvB[t]; mm[t] = mmB[t]; }

        // issue next chunk's loads (unconditional; covers compute below)
        const int dn = d + 4;
        avB = *(const v2f*)(aP + dn);
#pragma unroll
        for (int t = 0; t < NTILE; ++t) {
            vvB[t] = *(const v2f*)(vP + dn + t * (16 * DIM));
            mmB[t] = *(const v2f*)(mP + dn + t * (16 * DIM));
        }

        // compute current chunk
        v2f a2 = av * av;
        nsq += a2.x + a2.y;
#pragma unroll
        for (int t = 0; t < NTILE; ++t) {
            v2f r;  r.x = rcp_nr(vv[t].x); r.y = rcp_nr(vv[t].y);
            v2f bv = -0.5f * r;           // B frag: -0.5/v
            v2f bm = mm[t] * r;           // B frag:  m/v
            accLP[t] = __builtin_amdgcn_wmma_f32_16x16x4_f32(
                false, a2, false, bv, (short)0, accLP[t], false, false);
            accLP[t] = __builtin_amdgcn_wmma_f32_16x16x4_f32(
                false, av, false, bm, (short)0, accLP[t], false, false);
            accG3[t] = __builtin_amdgcn_wmma_f32_16x16x4_f32(
                false, av, false, mm[t], (short)0, accG3[t], false, false);
        }
    }

    // ---- pipeline epilogue: final chunk from buffers ----
    {
        v2f av = avB;
        v2f a2 = av * av;
        nsq += a2.x + a2.y;
#pragma unroll
        for (int t = 0; t < NTILE; ++t) {
            v2f r;  r.x = rcp_nr(vvB[t].x); r.y = rcp_nr(vvB[t].y);
            v2f bv = -0.5f * r;
            v2f bm = mmB[t] * r;
            accLP[t] = __builtin_amdgcn_wmma_f32_16x16x4_f32(
                false, a2, false, bv, (short)0, accLP[t], false, false);
            accLP[t] = __builtin_amdgcn_wmma_f32_16x16x4_f32(
                false, av, false, bm, (short)0, accLP[t], false, false);
            accG3[t] = __builtin_amdgcn_wmma_f32_16x16x4_f32(
                false, av, false, mmB[t], (short)0, accG3[t], false, false);
        }
    }

    // full ||a||^2 per row: lane r and lane r+16 hold complementary halves
    nsq += __shfl_xor(nsq, 16, 32);

    __shared__ float s_sum[16];
    __shared__ int   s_col[16];
    __shared__ float s_g3[16];
    __shared__ float s_nsq[16];
    if (lane < 16) s_nsq[lane] = nsq;

    // x = logp / TEMP, in place
#pragma unroll
    for (int t = 0; t < NTILE; ++t) {
        float ck = c_in[t * 16 + col];
        accLP[t] = (accLP[t] - 0.5f * ck) * TEMP_INV;
    }

    // per output row j (row index = j + 8*hi): max, argmax (+G3 value), sum(exp)
#pragma unroll
    for (int j = 0; j < 8; ++j) {
        float bval = accLP[0][j];
        int   bcol = col;
        float bg3  = accG3[0][j];
#pragma unroll
        for (int t = 1; t < NTILE; ++t) {
            if (accLP[t][j] > bval) {
                bval = accLP[t][j]; bcol = t * 16 + col; bg3 = accG3[t][j];
            }
        }
#pragma unroll
        for (int off = 1; off < 16; off <<= 1) {      // stays within 16-lane half
            float pv = __shfl_xor(bval, off, 32);
            int   pc = __shfl_xor(bcol, off, 32);
            float pg = __shfl_xor(bg3,  off, 32);
            if (pv > bval || (pv == bval && pc < bcol)) { bval = pv; bcol = pc; bg3 = pg; }
        }
        float se = 0.0f;                               // sum exp(x - xmax)
#pragma unroll
        for (int t = 0; t < NTILE; ++t) se += __expf(accLP[t][j] - bval);
#pragma unroll
        for (int off = 1; off < 16; off <<= 1) se += __shfl_xor(se, off, 32);

        if (col == 0) {
            int row = j + hi * 8;
            s_sum[row] = se; s_col[row] = bcol; s_g3[row] = bg3;
        }
    }
    __syncthreads();

    float contrib = 0.0f;
    if (lane < 16) {
        float se = s_sum[lane];
        int   kk = s_col[lane];
        float g3 = s_g3[lane];
        float na = sqrtf(s_nsq[lane]);
        float nm = mnorm_in[kk];
        float cs = g3 / (fmaxf(na, EPSN) * fmaxf(nm, EPSN));
        // max_prob = 1/se ; fold mean (1/N) in here
        contrib = PULL * (1.0f - cs) * (1.0f / se) / (float)N;
    }
#pragma unroll
    for (int off = 16; off > 0; off >>= 1) contrib += __shfl_xor(contrib, off, 32);
    if (lane == 0) partial[blockIdx.x] = contrib;
}

// ---------------------------------------------------------------------------
// Kernel 3: deterministic final reduction of per-block partials
// ---------------------------------------------------------------------------
__global__ __launch_bounds__(32)
void reduce_partials(const float* __restrict__ partial, float* __restrict__ out,
                     int nblocks) {
    const int lane = threadIdx.x;
    float s = 0.0f;
    for (int i = lane; i < nblocks; i += 32) s += partial[i];
#pragma unroll
    for (int off = 16; off > 0; off >>= 1) s += __shfl_xor(s, off, 32);
    if (lane == 0) out[0] = s;
}

extern "C" void kernel_launch(void* const* d_in, const int* in_sizes, int n_in,
                              void* d_out, int out_size, void* d_ws, size_t ws_size,
                              hipStream_t stream) {
    const float* a     = (const float*)d_in[0];   // [N, 512]
    const float* means = (const float*)d_in[1];   // [128, 512]
    const float* vars  = (const float*)d_in[2];   // [128, 512]
    // d_in[3] (pseudo_label) is unused by the reference loss.

    const int N       = in_sizes[0] / DIM;        // 8192
    const int nblocks = N / 16;                   // 512

    float* ws      = (float*)d_ws;
    float* partial = ws;                          // [nblocks]
    float* c_buf   = ws + nblocks;                // [KPROTO]
    float* mnorm   = c_buf + KPROTO;              // [KPROTO]

    proto_stats_kernel<<<KPROTO, 32, 0, stream>>>(means, vars, c_buf, mnorm);
    gauss_posterior_main<<<nblocks, 32, 0, stream>>>(a, means, vars, c_buf, mnorm,
                                                     partial, N);
    reduce_partials<<<1, 32, 0, stream>>>(partial, (float*)d_out, nblocks);
}